// LSTM_FB1_9320079033017
// MI455X (gfx1250) — compile-verified
//
#include <hip/hip_runtime.h>
#include <hip/hip_bf16.h>

typedef float v2f __attribute__((ext_vector_type(2)));
typedef float v8f __attribute__((ext_vector_type(8)));

#define B_TOTAL 65536
#define IN 128
#define HD 128
#define ROWS_PER_WG 128      // 8 waves x 16 rows
#define WSTRIDE 260          // K=256 + 4 pad DWORDs -> lanes hit distinct LDS banks

__device__ __forceinline__ float fast_sigmoid(float x) {
    return 1.0f / (1.0f + __expf(-x));
}
__device__ __forceinline__ float fast_tanh(float x) {
    // tanh(x) = 1 - 2/(e^{2x}+1)
    return 1.0f - 2.0f / (__expf(2.0f * x) + 1.0f);
}

__global__ __launch_bounds__(256, 2)
void lstm_fused_wmma(const float* __restrict__ x,
                     const float* __restrict__ h_prev,
                     const float* __restrict__ c_prev,
                     const float* __restrict__ Wx,
                     const float* __restrict__ Wh,
                     const float* __restrict__ bx,
                     const float* __restrict__ bh,
                     const float* __restrict__ bgate,
                     float* __restrict__ out)
{
    // Weight slice for this workgroup's 16 output columns:
    // wsl[(g*16+n)*WSTRIDE + k], k<128 from Wx[g][col][k], k>=128 from Wh[g][col][k-128]
    __shared__ float wsl[4 * 16 * WSTRIDE];

    const int tid    = threadIdx.x;
    const int wave   = tid >> 5;
    const int lane   = tid & 31;
    const int cTile  = blockIdx.y;      // 0..7  (16 cols each)
    const int rowBlk = blockIdx.x;      // 0..511

    // ---- stage weights (64 KB) into LDS: 4096 float4 loads across 256 threads ----
    for (int t = tid; t < 4096; t += 256) {
        int rowIdx = t >> 6;            // 0..63 == g*16 + n
        int q      = t & 63;            // float4 index within 256-float K row
        int g      = rowIdx >> 4;
        int n      = rowIdx & 15;
        int col    = cTile * 16 + n;
        int k4     = q * 4;
        const float* src = (k4 < 128)
            ? (Wx + ((size_t)(g * HD + col) * IN + k4))
            : (Wh + ((size_t)(g * HD + col) * HD + (k4 - 128)));
        float4 v = *(const float4*)src;                 // global_load_b128
        *(float4*)(&wsl[rowIdx * WSTRIDE + k4]) = v;    // ds_store_b128 (16B aligned)
    }
    __syncthreads();

    const int m  = lane & 15;           // A row / B col / D col within tile
    const int hi = lane >> 4;           // K-pair selector for A/B fragments
    const int rowBase = rowBlk * ROWS_PER_WG + wave * 16;

    const float* xrow = x      + (size_t)(rowBase + m) * IN;
    const float* hrow = h_prev + (size_t)(rowBase + m) * HD;

    // per-lane base pointers into the LDS weight rows for this lane's column
    const float* w0 = &wsl[(0 * 16 + m) * WSTRIDE];
    const float* w1 = &wsl[(1 * 16 + m) * WSTRIDE];
    const float* w2 = &wsl[(2 * 16 + m) * WSTRIDE];
    const float* w3 = &wsl[(3 * 16 + m) * WSTRIDE];

    v8f acc0 = {}, acc1 = {}, acc2 = {}, acc3 = {};

    // ---- x half: K = 0..127 ----
    #pragma unroll 4
    for (int kk = 0; kk < 128; kk += 4) {
        const int k0 = kk + 2 * hi;
        v2f a  = *(const v2f*)(xrow + k0);   // A frag: (m, k0), (m, k0+1)
        v2f b0 = *(const v2f*)(w0 + k0);     // B frag: (k0, n), (k0+1, n)
        v2f b1 = *(const v2f*)(w1 + k0);
        v2f b2 = *(const v2f*)(w2 + k0);
        v2f b3 = *(const v2f*)(w3 + k0);
        acc0 = __builtin_amdgcn_wmma_f32_16x16x4_f32(false, a, false, b0, (short)0, acc0, false, false);
        acc1 = __builtin_amdgcn_wmma_f32_16x16x4_f32(false, a, false, b1, (short)0, acc1, false, false);
        acc2 = __builtin_amdgcn_wmma_f32_16x16x4_f32(false, a, false, b2, (short)0, acc2, false, false);
        acc3 = __builtin_amdgcn_wmma_f32_16x16x4_f32(false, a, false, b3, (short)0, acc3, false, false);
    }
    // ---- h half: K = 128..255 ----
    #pragma unroll 4
    for (int kk = 0; kk < 128; kk += 4) {
        const int k0 = kk + 2 * hi;
        v2f a  = *(const v2f*)(hrow + k0);
        v2f b0 = *(const v2f*)(w0 + 128 + k0);
        v2f b1 = *(const v2f*)(w1 + 128 + k0);
        v2f b2 = *(const v2f*)(w2 + 128 + k0);
        v2f b3 = *(const v2f*)(w3 + 128 + k0);
        acc0 = __builtin_amdgcn_wmma_f32_16x16x4_f32(false, a, false, b0, (short)0, acc0, false, false);
        acc1 = __builtin_amdgcn_wmma_f32_16x16x4_f32(false, a, false, b1, (short)0, acc1, false, false);
        acc2 = __builtin_amdgcn_wmma_f32_16x16x4_f32(false, a, false, b2, (short)0, acc2, false, false);
        acc3 = __builtin_amdgcn_wmma_f32_16x16x4_f32(false, a, false, b3, (short)0, acc3, false, false);
    }

    // ---- fused gate epilogue ----
    const int col = cTile * 16 + m;     // D-matrix: lane&15 is the N index
    const float zb0 = bx[0 * HD + col] + bh[0 * HD + col];
    const float zb1 = bx[1 * HD + col] + bh[1 * HD + col];
    const float zb2 = bx[2 * HD + col] + bh[2 * HD + col];
    const float zb3 = bx[3 * HD + col] + bh[3 * HD + col];
    const float bg0 = bgate[0 * HD + col];
    const float bg1 = bgate[1 * HD + col];
    const float bg3 = bgate[3 * HD + col];

    const size_t outHalf = (size_t)B_TOTAL * HD;
    #pragma unroll
    for (int v = 0; v < 8; ++v) {
        const int row = rowBase + v + 8 * hi;   // D-matrix: VGPR v, lane-half hi
        const size_t idx = (size_t)row * HD + col;
        const float zf = acc0[v] + zb0;
        const float zi = acc1[v] + zb1;
        const float zc = acc2[v] + zb2;
        const float zo = acc3[v] + zb3;
        const float fg = fast_sigmoid(2.0f * zf + bg0);
        const float ig = fast_sigmoid(2.0f * zi + bg1);
        const float og = fast_sigmoid(2.0f * zo + bg3);
        const float cp = c_prev[idx];
        const float cn = fast_tanh(zc) * ig + fg * cp;
        const float hn = og * fast_tanh(cn);
        out[idx]           = hn;   // h_next
        out[outHalf + idx] = cn;   // c_next
    }
}

extern "C" void kernel_launch(void* const* d_in, const int* in_sizes, int n_in,
                              void* d_out, int out_size, void* d_ws, size_t ws_size,
                              hipStream_t stream) {
    const float* x      = (const float*)d_in[0];
    const float* h_prev = (const float*)d_in[1];
    const float* c_prev = (const float*)d_in[2];
    const float* Wx     = (const float*)d_in[3];
    const float* Wh     = (const float*)d_in[4];
    const float* bx     = (const float*)d_in[5];
    const float* bh     = (const float*)d_in[6];
    const float* bgate  = (const float*)d_in[7];
    float* out          = (float*)d_out;

    dim3 grid(B_TOTAL / ROWS_PER_WG, HD / 16);   // 512 x 8
    lstm_fused_wmma<<<grid, 256, 0, stream>>>(x, h_prev, c_prev, Wx, Wh, bx, bh, bgate, out);
}